// AngularDescriptor_7249904796077
// MI455X (gfx1250) — compile-verified
//
#include <hip/hip_runtime.h>
#include <hip/hip_bf16.h>
#include <math.h>

// CDNA5 / gfx1250. wave32 only. WMMA f32 16x16x4 keeps full f32 precision.
typedef float v2f __attribute__((ext_vector_type(2)));
typedef float v8f __attribute__((ext_vector_type(8)));

#define N_TYPES_   3
#define N_DESC_    8
#define K_MAX_     8
#define MAX_NN_    40
#define NBASIS_    24
#define RC_        4.0f
#define PI_        3.14159265358979323846f

#define WAVES_PER_BLOCK 4
#define BLOCK_THREADS   (WAVES_PER_BLOCK * 32)
#define CTAB_ELEMS      (N_TYPES_ * N_TYPES_ * N_DESC_ * K_MAX_) // 576

__global__ __launch_bounds__(BLOCK_THREADS)
void angdesc_kernel(const int* __restrict__ types,
                    const float* __restrict__ positions,
                    const int* __restrict__ nbrs,
                    const float* __restrict__ offsets,
                    const float* __restrict__ c_table,
                    float* __restrict__ out,
                    int n_atoms)
{
    __shared__ float ldsC[CTAB_ELEMS];                                 // 2.25 KB
    __shared__ float ldsGn[WAVES_PER_BLOCK][MAX_NN_ * N_DESC_];        // 4*1.25 KB
    __shared__ float ldsBs[WAVES_PER_BLOCK][MAX_NN_ * NBASIS_];        // 4*3.75 KB
    __shared__ float ldsS [WAVES_PER_BLOCK][N_DESC_ * NBASIS_];        // 4*0.75 KB

    const int tid  = threadIdx.x;
    const int lane = tid & 31;
    const int wave = tid >> 5;
    const int atom = blockIdx.x * WAVES_PER_BLOCK + wave;

    // Stage the small type-pair coefficient table in LDS (shared by block).
    for (int i = tid; i < CTAB_ELEMS; i += BLOCK_THREADS)
        ldsC[i] = c_table[i];
    __syncthreads();

    float* gn = ldsGn[wave];
    float* bs = ldsBs[wave];
    float* sm = ldsS[wave];

    // ---------------- Phase 1: per-pair radial (gn) and angular (basis) ----
    if (atom < n_atoms) {
        const int   ti  = types[atom];
        const float pix = positions[atom * 3 + 0];
        const float piy = positions[atom * 3 + 1];
        const float piz = positions[atom * 3 + 2];

        #pragma unroll
        for (int rep = 0; rep < 2; ++rep) {
            const int j = lane + rep * 32;
            if (j < MAX_NN_) {
                const int  nb   = nbrs[atom * MAX_NN_ + j];
                const bool mask = (nb != -1);
                const int  safe = mask ? nb : 0;

                float rx = positions[safe * 3 + 0] + offsets[(atom * MAX_NN_ + j) * 3 + 0] - pix;
                float ry = positions[safe * 3 + 1] + offsets[(atom * MAX_NN_ + j) * 3 + 1] - piy;
                float rz = positions[safe * 3 + 2] + offsets[(atom * MAX_NN_ + j) * 3 + 2] - piz;
                if (!mask) { rx = 0.f; ry = 0.f; rz = 0.f; }

                const float dist = sqrtf(rx * rx + ry * ry + rz * rz);
                const float inv  = 1.0f / fmaxf(dist, 1e-12f);
                const float x = rx * inv, y = ry * inv, z = rz * inv; // masked -> 0

                // Chebyshev radial basis with cosine cutoff.
                const float fcv = (dist < RC_) ? (0.5f * cosf(PI_ * dist * (1.0f / RC_)) + 0.5f) : 0.f;
                float xc = dist * (1.0f / RC_) - 1.0f;
                xc = 2.f * xc * xc - 1.f;
                float f[K_MAX_];
                {
                    float t0 = 1.f, t1 = xc;
                    f[0] = (t0 + 1.f) * 0.5f * fcv;
                    f[1] = (t1 + 1.f) * 0.5f * fcv;
                    #pragma unroll
                    for (int k = 2; k < K_MAX_; ++k) {
                        const float tk = 2.f * xc * t1 - t0;
                        t0 = t1; t1 = tk;
                        f[k] = (tk + 1.f) * 0.5f * fcv;
                    }
                }

                // gn[n] = c_table[ti,tj,n,:] . f[:]
                const int tj = types[safe];
                const float* cb = &ldsC[(ti * N_TYPES_ + tj) * N_DESC_ * K_MAX_];
                #pragma unroll
                for (int n = 0; n < N_DESC_; ++n) {
                    float acc = 0.f;
                    #pragma unroll
                    for (int k = 0; k < K_MAX_; ++k)
                        acc += cb[n * K_MAX_ + k] * f[k];
                    gn[j * N_DESC_ + n] = acc;
                }

                // Angular basis (expanded Z_COEFF_DATA recurrences), masked.
                const float m  = mask ? 1.f : 0.f;
                const float z2 = z * z, z3 = z2 * z, z4 = z3 * z;
                const float xr2 = x * x - y * y,     xi2 = 2.f * x * y;
                const float xr3 = xr2 * x - xi2 * y, xi3 = xr2 * y + xi2 * x;
                const float xr4 = xr3 * x - xi3 * y, xi4 = xr3 * y + xi3 * x;
                float b[NBASIS_];
                b[0] = z; b[1] = x; b[2] = y;                                   // L=1
                b[3] = 3.f * z2 - 1.f; b[4] = z * x; b[5] = z * y;              // L=2
                b[6] = xr2; b[7] = xi2;
                b[8] = 5.f * z3 - 3.f * z;                                      // L=3
                { const float t = 5.f * z2 - 1.f; b[9] = t * x; b[10] = t * y; }
                b[11] = z * xr2; b[12] = z * xi2; b[13] = xr3; b[14] = xi3;
                b[15] = 35.f * z4 - 30.f * z2 + 3.f;                            // L=4
                { const float t = 7.f * z3 - 3.f * z; b[16] = t * x; b[17] = t * y; }
                { const float t = 7.f * z2 - 1.f;     b[18] = t * xr2; b[19] = t * xi2; }
                b[20] = z * xr3; b[21] = z * xi3; b[22] = xr4; b[23] = xi4;
                #pragma unroll
                for (int a = 0; a < NBASIS_; ++a)
                    bs[j * NBASIS_ + a] = b[a] * m;
            }
        }
    }
    __syncthreads();

    // ---------------- Phase 2: s(8x24) = gn^T(8x40) x basis(40x24) via WMMA -
    // V_WMMA_F32_16X16X4_F32, K chunked by 4 (10 chunks), N split 16 + 8.
    // A layout: lanes 0-15 -> M=lane, v[0]=K0,v[1]=K1; lanes 16-31 -> K2/K3.
    // B layout (symmetric): lanes 0-15 -> N=lane, v[0]=K0,v[1]=K1; 16-31 -> K2/K3.
    if (atom < n_atoms) {
        v8f acc0 = {}; // s columns 0..15
        v8f acc1 = {}; // s columns 16..23 (cols 8..15 of this tile are zero)
        const int  half   = lane >> 4;
        const int  lrow   = lane & 15;
        const int  r8     = lrow & 7;
        const bool lowrow = (lrow < 8);

        #pragma unroll
        for (int c = 0; c < MAX_NN_ / 4; ++c) {
            const int j0 = 4 * c + 2 * half; // K row for fragment element 0
            const int j1 = j0 + 1;           // K row for fragment element 1

            v2f afrag;
            { float a0 = gn[j0 * N_DESC_ + r8];
              float a1 = gn[j1 * N_DESC_ + r8];
              afrag[0] = lowrow ? a0 : 0.f;    // rows 8..15 of A are zero pad
              afrag[1] = lowrow ? a1 : 0.f; }

            v2f bfrag0;
            bfrag0[0] = bs[j0 * NBASIS_ + lrow];
            bfrag0[1] = bs[j1 * NBASIS_ + lrow];

            v2f bfrag1;
            { float u0 = bs[j0 * NBASIS_ + 16 + r8];
              float u1 = bs[j1 * NBASIS_ + 16 + r8];
              bfrag1[0] = lowrow ? u0 : 0.f;   // cols 24..31 zero pad
              bfrag1[1] = lowrow ? u1 : 0.f; }

            acc0 = __builtin_amdgcn_wmma_f32_16x16x4_f32(
                false, afrag, false, bfrag0, (short)0, acc0, false, false);
            acc1 = __builtin_amdgcn_wmma_f32_16x16x4_f32(
                false, afrag, false, bfrag1, (short)0, acc1, false, false);
        }

        // D layout: VGPR d, lanes 0-15 -> (M=d, N=lane); lanes 16-31 -> M=8+d (pad).
        #pragma unroll
        for (int d = 0; d < N_DESC_; ++d) {
            if (lane < 16) sm[d * NBASIS_ + lane]      = acc0[d];
            if (lane < 8)  sm[d * NBASIS_ + 16 + lane] = acc1[d];
        }
    }
    __syncthreads();

    // ---------------- Phase 3: q invariants per descriptor channel ----------
    if (atom < n_atoms && lane < N_DESC_) {
        float s[NBASIS_];
        #pragma unroll
        for (int a = 0; a < NBASIS_; ++a)
            s[a] = sm[lane * NBASIS_ + a];

        const float C3B24[NBASIS_] = {
            0.238732414637843f, 0.119366207318922f, 0.119366207318922f,
            0.099471839432435f, 0.596831036594608f, 0.596831036594608f,
            0.149207759148652f, 0.149207759148652f,
            0.139260575205408f, 0.104445431404056f, 0.104445431404056f,
            1.044454314040563f, 1.044454314040563f, 0.174075719006761f,
            0.174075719006761f,
            0.011190581936149f, 0.223811638722978f, 0.223811638722978f,
            0.111905819361489f, 0.111905819361489f, 1.566681471060845f,
            1.566681471060845f, 0.195835183882606f, 0.195835183882606f };
        const int starts[4] = {0, 3, 8, 15};
        const int nts[4]    = {3, 5, 7, 9};

        float qv[6];
        #pragma unroll
        for (int L = 0; L < 4; ++L) {
            float q = C3B24[starts[L]] * s[starts[L]] * s[starts[L]];
            #pragma unroll
            for (int t = 1; t < 9; ++t) {
                if (t < nts[L]) {
                    const int a = starts[L] + t;
                    q += 2.f * C3B24[a] * s[a] * s[a];
                }
            }
            qv[L] = q;
        }
        // 4-body
        {
            const float s3 = s[3], s4 = s[4], s5 = s[5], s6 = s[6], s7 = s[7];
            qv[4] = -0.007499480826664f * s3 * s3 * s3
                  + -0.134990654879954f * s3 * (s4 * s4 + s5 * s5)
                  +  0.067495327439977f * s3 * (s6 * s6 + s7 * s7)
                  +  0.404971964639861f * s6 * (s5 * s5 - s4 * s4)
                  + -0.809943929279723f * s4 * s5 * s7;
        }
        // 5-body
        {
            const float aa = s[0] * s[0];
            const float bb = s[1] * s[1] + s[2] * s[2];
            qv[5] = 0.026596810706114f * aa * aa
                  + 0.053193621412227f * aa * bb
                  + 0.026596810706114f * bb * bb;
        }

        float* o = out + (size_t)(atom * N_DESC_ + lane) * 6;
        #pragma unroll
        for (int t = 0; t < 6; ++t) o[t] = qv[t];
    }
}

extern "C" void kernel_launch(void* const* d_in, const int* in_sizes, int n_in,
                              void* d_out, int out_size, void* d_ws, size_t ws_size,
                              hipStream_t stream) {
    (void)n_in; (void)out_size; (void)d_ws; (void)ws_size;
    const int*   types     = (const int*)d_in[0];
    const float* positions = (const float*)d_in[1];
    const int*   nbrs      = (const int*)d_in[2];
    const float* offsets   = (const float*)d_in[3];
    const float* c_table   = (const float*)d_in[4];
    float*       out       = (float*)d_out;

    const int n_atoms = in_sizes[0];
    const int grid = (n_atoms + WAVES_PER_BLOCK - 1) / WAVES_PER_BLOCK;
    angdesc_kernel<<<grid, BLOCK_THREADS, 0, stream>>>(
        types, positions, nbrs, offsets, c_table, out, n_atoms);
}